// GConvNetBigGraph_78554951844406
// MI455X (gfx1250) — compile-verified
//
#include <hip/hip_runtime.h>
#include <hip/hip_bf16.h>
#include <math.h>

typedef __attribute__((ext_vector_type(2))) float v2f;
typedef __attribute__((ext_vector_type(8))) float v8f;

#define IN_F 30
#define H1   30
#define H1P  32    // padded row stride for h/agg (128B rows)
#define H2   10
#define OUTF 4
#define NPG  40

// ---------------- zero fill ----------------
__global__ void zero_f32(float* __restrict__ p, long long n) {
  long long i = (long long)blockIdx.x * blockDim.x + threadIdx.x;
  long long stride = (long long)gridDim.x * blockDim.x;
  for (; i < n; i += stride) p[i] = 0.0f;
}

// ---------------- pack W into zero-padded, fragment-ordered buffer ------------
// Wpp[kbh*64 + n*2 + p] = W[2*kbh + p][n]  (zero outside 30x30), kbh=0..15, n=0..31
// A B-fragment for (kb, half) at column n is then one aligned 8B load.
__global__ void prep_w(const float* __restrict__ W, float* __restrict__ Wpp) {
  int idx = blockIdx.x * blockDim.x + threadIdx.x;
  if (idx < 1024) {
    int kbh = idx >> 6;
    int rem = idx & 63;
    int n = rem >> 1;
    int p = rem & 1;
    int k = kbh * 2 + p;
    Wpp[idx] = (k < IN_F && n < H1) ? W[k * H1 + n] : 0.0f;
  }
}

// ---------------- degree computation (L2-resident atomics) ----------------
__global__ void degree_kernel(const int* __restrict__ src, const int* __restrict__ dst,
                              float* __restrict__ deg_out, float* __restrict__ deg_in,
                              int n_edges) {
  int e = blockIdx.x * blockDim.x + threadIdx.x;
  if (e < n_edges) {
    atomicAdd(&deg_out[src[e]], 1.0f);
    atomicAdd(&deg_in[dst[e]], 1.0f);
  }
}

// ---------------- GEMM1: h = (feat * norm_src) @ W  via V_WMMA_F32_16X16X4_F32 ----
// One wave per 16-row tile. N=30 -> two 16-col WMMA tiles stored into stride-32
// padded h (cols 30/31 are dead). K=30 padded to 32 (8 steps). All loads/stores
// unconditional: no exec-divergent predication in the hot loop.
__global__ void gemm1_wmma(const float* __restrict__ feat, const float* __restrict__ Wpp,
                           const float* __restrict__ deg_out, float* __restrict__ h,
                           int n_nodes) {
  int wid  = (blockIdx.x * blockDim.x + threadIdx.x) >> 5;
  int lane = threadIdx.x & 31;
  int m0   = wid * 16;
  if (m0 >= n_nodes) return;           // wave-uniform: EXEC all-1 for WMMA below

  int half = lane >> 4;                // 0: lanes 0-15, 1: lanes 16-31
  int l16  = lane & 15;
  int row  = m0 + l16;
  if (row >= n_nodes) row = n_nodes - 1;

  float ns = rsqrtf(fmaxf(deg_out[row], 1.0f));
  const float* frow = feat + (long long)row * IN_F;

  v8f c0 = {};
  v8f c1 = {};
  #pragma unroll
  for (int kb = 0; kb < 8; ++kb) {
    int k0  = kb * 4 + half * 2;       // A 16x4 f32: lanes0-15 K={k,k+1}, lanes16-31 K={k+2,k+3}
    int kbh = kb * 2 + half;

    v2f a;
    if (kb < 7) {
      a = *(const v2f*)(frow + k0);                 // aligned 8B load, always valid
    } else {
      v2f t = *(const v2f*)(frow + 28);             // unconditional; zero via data select
      a.x = half ? 0.0f : t.x;
      a.y = half ? 0.0f : t.y;
    }
    a.x *= ns;
    a.y *= ns;

    v2f b0 = *(const v2f*)(Wpp + kbh * 64 + l16 * 2);         // cols 0-15
    v2f b1 = *(const v2f*)(Wpp + kbh * 64 + (16 + l16) * 2);  // cols 16-31 (zero-padded)

    c0 = __builtin_amdgcn_wmma_f32_16x16x4_f32(false, a, false, b0, (short)0, c0, false, false);
    c1 = __builtin_amdgcn_wmma_f32_16x16x4_f32(false, a, false, b1, (short)0, c1, false, false);
  }

  // D 16x16 f32 layout: VGPR r -> M = r (lanes 0-15) / r+8 (lanes 16-31), N = l16
  if (m0 + 16 <= n_nodes) {            // wave-uniform fast path: unconditional stores
    #pragma unroll
    for (int r = 0; r < 8; ++r) {
      long long orow = m0 + r + half * 8;
      h[orow * H1P + l16]      = c0[r];
      h[orow * H1P + 16 + l16] = c1[r];
    }
  } else {
    #pragma unroll
    for (int r = 0; r < 8; ++r) {
      long long orow = m0 + r + half * 8;
      if (orow < n_nodes) {
        h[orow * H1P + l16]      = c0[r];
        h[orow * H1P + 16 + l16] = c1[r];
      }
    }
  }
}

// ---------------- message passing: agg[dst] += h[src] (lane = feature) ----------
__global__ void scatter_kernel(const int* __restrict__ src, const int* __restrict__ dst,
                               const float* __restrict__ h, float* __restrict__ agg,
                               int n_edges) {
  long long t = (long long)blockIdx.x * blockDim.x + threadIdx.x;
  int e = (int)(t >> 5);
  int f = (int)(t & 31);
  if (e < n_edges && f < H1) {
    int s = src[e];
    int d = dst[e];
    atomicAdd(&agg[(long long)d * H1P + f], h[(long long)s * H1P + f]);
  }
}

// ---------------- fused: norm+bias, per-graph max-pool, MLP head, sigmoid ------
// One wave per graph (40 contiguous nodes). Lane = feature for pooling, then
// __shfl broadcasts for the tiny 30->10->4 head.
__global__ void pool_head_kernel(const float* __restrict__ agg, const float* __restrict__ deg_in,
                                 const float* __restrict__ b,  const float* __restrict__ W2,
                                 const float* __restrict__ b2, const float* __restrict__ W3,
                                 const float* __restrict__ b3, float* __restrict__ out,
                                 int n_graphs) {
  int g    = (blockIdx.x * blockDim.x + threadIdx.x) >> 5;
  int lane = threadIdx.x & 31;
  if (g >= n_graphs) return;

  float m = -3.402823466e38f;
  for (int i = 0; i < NPG; ++i) {
    int n = g * NPG + i;
    float nd = rsqrtf(fmaxf(deg_in[n], 1.0f));
    if (lane < H1) {
      float v = fmaf(agg[(long long)n * H1P + lane], nd, b[lane]);
      m = fmaxf(m, v);
    }
  }

  // z_j = relu(sum_f pooled[f] * W2[f][j] + b2[j]); pooled[f] lives in lane f
  int j2 = (lane < H2) ? lane : 0;
  float z = b2[j2];
  #pragma unroll
  for (int f = 0; f < H1; ++f) {
    float v = __shfl(m, f, 32);
    z = fmaf(v, W2[f * H2 + j2], z);
  }
  z = fmaxf(z, 0.0f);

  // o_j = sigmoid(sum_i z_i * W3[i][j] + b3[j]); z_i lives in lane i (i<10)
  int j3 = (lane < OUTF) ? lane : 0;
  float o = b3[j3];
  #pragma unroll
  for (int i = 0; i < H2; ++i) {
    float zv = __shfl(z, i, 32);
    o = fmaf(zv, W3[i * OUTF + j3], o);
  }
  o = 1.0f / (1.0f + expf(-o));

  if (lane < OUTF) out[(long long)g * OUTF + lane] = o;
}

// ---------------- launch ----------------
extern "C" void kernel_launch(void* const* d_in, const int* in_sizes, int n_in,
                              void* d_out, int out_size, void* d_ws, size_t ws_size,
                              hipStream_t stream) {
  const float* feat = (const float*)d_in[0];
  const int*   src  = (const int*)d_in[1];
  const int*   dst  = (const int*)d_in[2];
  // d_in[3] segment_ids unused (graphs are contiguous runs of 40 nodes)
  // d_in[4] num_graphs (device scalar) unused; derived from sizes
  const float* W  = (const float*)d_in[5];
  const float* b  = (const float*)d_in[6];
  const float* W2 = (const float*)d_in[7];
  const float* b2 = (const float*)d_in[8];
  const float* W3 = (const float*)d_in[9];
  const float* b3 = (const float*)d_in[10];

  int n_nodes  = in_sizes[0] / IN_F;
  int n_edges  = in_sizes[1];
  int n_graphs = n_nodes / NPG;

  // workspace layout (floats): Wpp[1024] | deg_out[N] | deg_in[N] | h[N*32] | agg[N*32]
  float* ws      = (float*)d_ws;
  float* Wpp     = ws;
  float* deg_out = Wpp + 1024;
  float* deg_in  = deg_out + n_nodes;
  float* h       = deg_in + n_nodes;
  float* agg     = h + (long long)n_nodes * H1P;

  prep_w<<<4, 256, 0, stream>>>(W, Wpp);
  zero_f32<<<1024, 256, 0, stream>>>(deg_out, 2LL * n_nodes);                 // deg_out + deg_in contiguous
  zero_f32<<<8192, 256, 0, stream>>>(agg, (long long)n_nodes * H1P);

  degree_kernel<<<(n_edges + 255) / 256, 256, 0, stream>>>(src, dst, deg_out, deg_in, n_edges);

  int m_tiles = (n_nodes + 15) / 16;                                          // one wave per tile
  gemm1_wmma<<<(m_tiles + 7) / 8, 256, 0, stream>>>(feat, Wpp, deg_out, h, n_nodes);

  long long sc_threads = (long long)n_edges * 32;
  scatter_kernel<<<(unsigned)((sc_threads + 255) / 256), 256, 0, stream>>>(src, dst, h, agg, n_edges);

  pool_head_kernel<<<(n_graphs + 7) / 8, 256, 0, stream>>>(agg, deg_in, b, W2, b2, W3, b3,
                                                           (float*)d_out, n_graphs);
}